// GVPVectorField_49263274885950
// MI455X (gfx1250) — compile-verified
//
#include <hip/hip_runtime.h>
#include <math.h>

// ---------------------------------------------------------------------------
// GVP graph-network forward for MI455X (gfx1250).
// fp32 end-to-end; GEMMs run on V_WMMA_F32_16X16X4_F32 (exact fp32 matrix math
// vs. the JAX fp32 reference). Workload is gather/scatter bound (~3-4 GB edge
// traffic vs ~100 GFLOP), so fp32 WMMA + minimized data movement is the point.
// GEMM inner loops are branch-free: row/col indices are clamped instead of
// predicated (all M values are multiples of 16; mis-clamped lanes feed output
// rows/cols that the epilogue never stores), so EXEC is all-ones at every WMMA.
// ---------------------------------------------------------------------------

typedef __attribute__((ext_vector_type(2))) float v2f;
typedef __attribute__((ext_vector_type(8))) float v8f;

#define ACT_NONE    0
#define ACT_SILU    1
#define ACT_SIGMOID 2

__device__ __forceinline__ float sigm_(float x) { return 1.0f / (1.0f + expf(-x)); }

__device__ __forceinline__ v8f wmma4_(v2f a, v2f b, v8f c) {
  return __builtin_amdgcn_wmma_f32_16x16x4_f32(false, a, false, b, (short)0, c,
                                               false, false);
}

// --------------------------- GEMM, N == 64 ---------------------------------
// One wave computes a 16x64 output tile: 4 accumulators share each A fragment.
__global__ void gvpvf_gemm_wmma64(const float* __restrict__ A0, int lda0, int K0,
                                  const float* __restrict__ A1, int lda1, int K1,
                                  const float* __restrict__ W, const float* __restrict__ bias,
                                  float* __restrict__ Y, int M, int act) {
  const int lane = threadIdx.x;          // 0..31 (wave32)
  const int mrow = lane & 15;
  const int hi   = lane >> 4;
  const int tileM = blockIdx.x * 16;
  int r = tileM + mrow;
  if (r >= M) r = M - 1;                 // clamp: lanes past M feed unstored rows
  const int ncol = mrow;                 // column within each 16-wide group

  v8f acc0 = {}, acc1 = {}, acc2 = {}, acc3 = {};

  const float* Aseg[2] = { A0 + (size_t)r * lda0, A1 + (size_t)r * lda1 };
  const int    Lseg[2] = { K0, K1 };
  int kb = 0;
  for (int s = 0; s < 2; ++s) {
    const float* __restrict__ Ar = Aseg[s];
    const int L  = Lseg[s];
    const int Lf = L & ~3;
    for (int k = 0; k < Lf; k += 4) {
      const int kk = k + 2 * hi;
      v2f a; a.x = Ar[kk]; a.y = Ar[kk + 1];
      const float* Wr0 = W + (size_t)(kb + kk) * 64 + ncol;
      const float* Wr1 = Wr0 + 64;
      v2f b0; b0.x = Wr0[0];  b0.y = Wr1[0];
      v2f b1; b1.x = Wr0[16]; b1.y = Wr1[16];
      v2f b2; b2.x = Wr0[32]; b2.y = Wr1[32];
      v2f b3; b3.x = Wr0[48]; b3.y = Wr1[48];
      acc0 = wmma4_(a, b0, acc0);
      acc1 = wmma4_(a, b1, acc1);
      acc2 = wmma4_(a, b2, acc2);
      acc3 = wmma4_(a, b3, acc3);
    }
    if (L & 3) {                          // masked tail (zero-padded rank-4)
      const int kk = Lf + 2 * hi;
      const bool v0 = (kk < L), v1 = (kk + 1 < L);
      v2f a; a.x = v0 ? Ar[kk] : 0.f; a.y = v1 ? Ar[kk + 1] : 0.f;
      const float* Wr0 = W + (size_t)(kb + kk) * 64 + ncol;
      const float* Wr1 = Wr0 + 64;
      v2f b0; b0.x = v0 ? Wr0[0]  : 0.f; b0.y = v1 ? Wr1[0]  : 0.f;
      v2f b1; b1.x = v0 ? Wr0[16] : 0.f; b1.y = v1 ? Wr1[16] : 0.f;
      v2f b2; b2.x = v0 ? Wr0[32] : 0.f; b2.y = v1 ? Wr1[32] : 0.f;
      v2f b3; b3.x = v0 ? Wr0[48] : 0.f; b3.y = v1 ? Wr1[48] : 0.f;
      acc0 = wmma4_(a, b0, acc0);
      acc1 = wmma4_(a, b1, acc1);
      acc2 = wmma4_(a, b2, acc2);
      acc3 = wmma4_(a, b3, acc3);
    }
    kb += L;
  }

  const float bv0 = bias[ncol];
  const float bv1 = bias[ncol + 16];
  const float bv2 = bias[ncol + 32];
  const float bv3 = bias[ncol + 48];
#pragma unroll
  for (int i = 0; i < 8; ++i) {
    const int row = tileM + i + 8 * hi;
    if (row < M) {
      float y0 = acc0[i] + bv0, y1 = acc1[i] + bv1;
      float y2 = acc2[i] + bv2, y3 = acc3[i] + bv3;
      if (act == ACT_SILU) {
        y0 *= sigm_(y0); y1 *= sigm_(y1); y2 *= sigm_(y2); y3 *= sigm_(y3);
      } else if (act == ACT_SIGMOID) {
        y0 = sigm_(y0); y1 = sigm_(y1); y2 = sigm_(y2); y3 = sigm_(y3);
      }
      float* yr = Y + (size_t)row * 64 + ncol;
      yr[0] = y0; yr[16] = y1; yr[32] = y2; yr[48] = y3;
    }
  }
}

// --------------------------- GEMM, generic N -------------------------------
__global__ void gvpvf_gemm_wmma(const float* __restrict__ A0, int lda0, int K0,
                                const float* __restrict__ A1, int lda1, int K1,
                                const float* __restrict__ W, const float* __restrict__ bias,
                                float* __restrict__ Y, int ldy,
                                int M, int N, int act) {
  const int lane = threadIdx.x;
  const int mrow = lane & 15;
  const int hi   = lane >> 4;
  const int tileM = blockIdx.x * 16;
  const int tileN = blockIdx.y * 16;
  int r = tileM + mrow; if (r >= M) r = M - 1;
  const int n = tileN + mrow;
  int nc = n; if (nc >= N) nc = N - 1;   // clamp: past-N lanes feed unstored cols

  v8f acc = {};
  const float* Aseg[2] = { A0 + (size_t)r * lda0, A1 + (size_t)r * lda1 };
  const int    Lseg[2] = { K0, K1 };
  int kb = 0;
  for (int s = 0; s < 2; ++s) {
    const float* __restrict__ Ar = Aseg[s];
    const int L  = Lseg[s];
    const int Lf = L & ~3;
    for (int k = 0; k < Lf; k += 4) {
      const int kk = k + 2 * hi;
      v2f a; a.x = Ar[kk]; a.y = Ar[kk + 1];
      v2f b; b.x = W[(size_t)(kb + kk) * N + nc];
             b.y = W[(size_t)(kb + kk + 1) * N + nc];
      acc = wmma4_(a, b, acc);
    }
    if (L & 3) {
      const int kk = Lf + 2 * hi;
      const bool v0 = (kk < L), v1 = (kk + 1 < L);
      v2f a; a.x = v0 ? Ar[kk] : 0.f; a.y = v1 ? Ar[kk + 1] : 0.f;
      v2f b; b.x = v0 ? W[(size_t)(kb + kk) * N + nc]     : 0.f;
             b.y = v1 ? W[(size_t)(kb + kk + 1) * N + nc] : 0.f;
      acc = wmma4_(a, b, acc);
    }
    kb += L;
  }

  if (n < N) {
    const float bv = bias ? bias[n] : 0.0f;
#pragma unroll
    for (int i = 0; i < 8; ++i) {
      const int row = tileM + i + 8 * hi;
      if (row < M) {
        float y = acc[i] + bv;
        if (act == ACT_SILU)         y = y * sigm_(y);
        else if (act == ACT_SIGMOID) y = sigm_(y);
        Y[(size_t)row * ldy + n] = y;
      }
    }
  }
}

// Per-row GVP vector stage: Vh = v @ Wh ; sh = ||Vh||_c ; Vu = Vh @ Wu.
template <int VIN, int H, int VOUT>
__global__ void gvpvf_gvp_vec(const float* __restrict__ mv,
                              const float* __restrict__ Wh,
                              const float* __restrict__ Wu,
                              float* __restrict__ sh,
                              float* __restrict__ Vu, int n) {
  const int i = blockIdx.x * blockDim.x + threadIdx.x;
  if (i >= n) return;
  float vr[VIN * 3];
  const float* v = mv + (size_t)i * (VIN * 3);
#pragma unroll
  for (int j = 0; j < VIN * 3; ++j) vr[j] = v[j];
  float vh[H * 3];
#pragma unroll
  for (int hh = 0; hh < H; ++hh) {
    float a0 = 0.f, a1 = 0.f, a2 = 0.f;
#pragma unroll
    for (int vi = 0; vi < VIN; ++vi) {
      const float w = Wh[vi * H + hh];
      a0 += w * vr[vi * 3 + 0];
      a1 += w * vr[vi * 3 + 1];
      a2 += w * vr[vi * 3 + 2];
    }
    vh[hh * 3 + 0] = a0; vh[hh * 3 + 1] = a1; vh[hh * 3 + 2] = a2;
    sh[(size_t)i * H + hh] = sqrtf(a0 * a0 + a1 * a1 + a2 * a2 + 1e-8f);
  }
  float* vu = Vu + (size_t)i * (VOUT * 3);
#pragma unroll
  for (int u = 0; u < VOUT; ++u) {
    float a0 = 0.f, a1 = 0.f, a2 = 0.f;
#pragma unroll
    for (int hh = 0; hh < H; ++hh) {
      const float w = Wu[hh * VOUT + u];
      a0 += w * vh[hh * 3 + 0];
      a1 += w * vh[hh * 3 + 1];
      a2 += w * vh[hh * 3 + 2];
    }
    vu[u * 3 + 0] = a0; vu[u * 3 + 1] = a1; vu[u * 3 + 2] = a2;
  }
}

// mv[i, u, c] = g[i, u] * Vu[i, u, c]
__global__ void gvpvf_gate_combine(const float* __restrict__ g,
                                   const float* __restrict__ Vu,
                                   float* __restrict__ mv, int n, int vout) {
  const size_t idx = (size_t)blockIdx.x * blockDim.x + threadIdx.x;
  if (idx >= (size_t)n * vout * 3) return;
  const size_t t = idx / 3;
  const int u = (int)(t % vout);
  const size_t row = t / vout;
  mv[idx] = g[row * vout + u] * Vu[idx];
}

// s0 = [a_t(10) | c_t(6) | t[nbi](1)]
__global__ void gvpvf_build_s0(const float* __restrict__ a_t, const float* __restrict__ c_t,
                               const float* __restrict__ t, const int* __restrict__ nbi,
                               float* __restrict__ s0, int n) {
  const int i = blockIdx.x * blockDim.x + threadIdx.x;
  if (i >= n) return;
  float* o = s0 + (size_t)i * 17;
#pragma unroll
  for (int j = 0; j < 10; ++j) o[j] = a_t[(size_t)i * 10 + j];
#pragma unroll
  for (int j = 0; j < 6; ++j) o[10 + j] = c_t[(size_t)i * 6 + j];
  o[16] = t[nbi[i]];
}

// LayerNorm(dim=64) with optional fused residual: y = LN(a + scale*add)*g + b
__global__ void gvpvf_ln64(const float* __restrict__ a, const float* __restrict__ add,
                           float scale, const float* __restrict__ gw,
                           const float* __restrict__ bw, float* __restrict__ out, int n) {
  const int i = blockIdx.x * blockDim.x + threadIdx.x;
  if (i >= n) return;
  float x[64];
  float m = 0.f;
#pragma unroll
  for (int j = 0; j < 64; ++j) {
    float v = a[(size_t)i * 64 + j];
    if (add) v += scale * add[(size_t)i * 64 + j];
    x[j] = v;
    m += v;
  }
  m *= (1.0f / 64.0f);
  float var = 0.f;
#pragma unroll
  for (int j = 0; j < 64; ++j) { const float d = x[j] - m; var += d * d; }
  var *= (1.0f / 64.0f);
  const float inv = rsqrtf(var + 1e-5f);
#pragma unroll
  for (int j = 0; j < 64; ++j)
    out[(size_t)i * 64 + j] = (x[j] - m) * inv * gw[j] + bw[j];
}

__global__ void gvpvf_distances(const float* __restrict__ x, const int* __restrict__ src,
                                const int* __restrict__ dst, float* __restrict__ xd,
                                float* __restrict__ rbf, int E_) {
  const int e = blockIdx.x * blockDim.x + threadIdx.x;
  if (e >= E_) return;
  const int s = src[e], d = dst[e];
  const float dx = x[(size_t)s * 3 + 0] - x[(size_t)d * 3 + 0];
  const float dy = x[(size_t)s * 3 + 1] - x[(size_t)d * 3 + 1];
  const float dz = x[(size_t)s * 3 + 2] - x[(size_t)d * 3 + 2];
  const float dij = sqrtf(dx * dx + dy * dy + dz * dz + 1e-8f) + 1e-8f;
  const float inv = 1.0f / dij;
  xd[(size_t)e * 3 + 0] = dx * inv;
  xd[(size_t)e * 3 + 1] = dy * inv;
  xd[(size_t)e * 3 + 2] = dz * inv;
  const float invsig = 16.0f / 20.0f;  // 1/sigma, sigma = 20/16
#pragma unroll
  for (int j = 0; j < 16; ++j) {
    const float mu = 20.0f * (float)j / 15.0f;
    const float z = (dij - mu) * invsig;
    rbf[(size_t)e * 16 + j] = expf(-z * z);
  }
}

// ms(208) = [s[src] | s[dst] | e | rbf] ; mv(17x3) = [v[src] ; x_diff]
__global__ void gvpvf_gather_msg(const float* __restrict__ s, const float* __restrict__ v,
                                 const float* __restrict__ e, const float* __restrict__ rbf,
                                 const float* __restrict__ xd, const int* __restrict__ src,
                                 const int* __restrict__ dst, float* __restrict__ ms,
                                 float* __restrict__ mv, int E_) {
  const int ei = blockIdx.x * blockDim.x + threadIdx.x;
  if (ei >= E_) return;
  const int si = src[ei], di = dst[ei];
  float* mo = ms + (size_t)ei * 208;
  const float* ss = s + (size_t)si * 64;
  const float* sd = s + (size_t)di * 64;
  const float* ee = e + (size_t)ei * 64;
#pragma unroll 4
  for (int j = 0; j < 64; ++j) { mo[j] = ss[j]; mo[64 + j] = sd[j]; mo[128 + j] = ee[j]; }
  const float* rb = rbf + (size_t)ei * 16;
#pragma unroll
  for (int j = 0; j < 16; ++j) mo[192 + j] = rb[j];
  float* vo = mv + (size_t)ei * 51;
  const float* vs = v + (size_t)si * 48;
#pragma unroll 4
  for (int j = 0; j < 48; ++j) vo[j] = vs[j];
  vo[48] = xd[(size_t)ei * 3 + 0];
  vo[49] = xd[(size_t)ei * 3 + 1];
  vo[50] = xd[(size_t)ei * 3 + 2];
}

// h(192) = [s[src] | s[dst] | e]
__global__ void gvpvf_gather_eupd(const float* __restrict__ s, const float* __restrict__ e,
                                  const int* __restrict__ src, const int* __restrict__ dst,
                                  float* __restrict__ h, int E_) {
  const int ei = blockIdx.x * blockDim.x + threadIdx.x;
  if (ei >= E_) return;
  const int si = src[ei], di = dst[ei];
  float* o = h + (size_t)ei * 192;
  const float* ss = s + (size_t)si * 64;
  const float* sd = s + (size_t)di * 64;
  const float* ee = e + (size_t)ei * 64;
#pragma unroll 4
  for (int j = 0; j < 64; ++j) { o[j] = ss[j]; o[64 + j] = sd[j]; o[128 + j] = ee[j]; }
}

__global__ void gvpvf_scatter_add(const float* __restrict__ data, const int* __restrict__ dst,
                                  float* __restrict__ out, int E_, int width) {
  const size_t idx = (size_t)blockIdx.x * blockDim.x + threadIdx.x;
  if (idx >= (size_t)E_ * width) return;
  const int e = (int)(idx / width);
  const int f = (int)(idx % width);
  atomicAdd(&out[(size_t)dst[e] * width + f], data[idx]);
}

__global__ void gvpvf_add_scaled(float* __restrict__ out, const float* __restrict__ a,
                                 float scale, size_t n) {
  const size_t i = (size_t)blockIdx.x * blockDim.x + threadIdx.x;
  if (i < n) out[i] += scale * a[i];
}

__global__ void gvpvf_pair_add(const float* __restrict__ e, float* __restrict__ pb,
                               size_t n, size_t off) {
  const size_t i = (size_t)blockIdx.x * blockDim.x + threadIdx.x;
  if (i < n) pb[i] = e[i] + e[i + off];
}

__global__ void gvpvf_head_split(const float* __restrict__ no, float* __restrict__ c_out,
                                 float* __restrict__ a_out, int n) {
  const int i = blockIdx.x * blockDim.x + threadIdx.x;
  if (i >= n) return;
  const float* r = no + (size_t)i * 16;
#pragma unroll
  for (int j = 0; j < 6; ++j) c_out[(size_t)i * 6 + j] = r[j];
#pragma unroll
  for (int j = 0; j < 10; ++j) a_out[(size_t)i * 10 + j] = r[6 + j];
}

__global__ void gvpvf_graph_accum(const float* __restrict__ x, const int* __restrict__ nbi,
                                  float* __restrict__ gs, int n) {
  const int i = blockIdx.x * blockDim.x + threadIdx.x;
  if (i >= n) return;
  const int g = nbi[i];
  atomicAdd(&gs[(size_t)g * 4 + 0], x[(size_t)i * 3 + 0]);
  atomicAdd(&gs[(size_t)g * 4 + 1], x[(size_t)i * 3 + 1]);
  atomicAdd(&gs[(size_t)g * 4 + 2], x[(size_t)i * 3 + 2]);
  atomicAdd(&gs[(size_t)g * 4 + 3], 1.0f);
}

__global__ void gvpvf_center_x(const float* __restrict__ x, const int* __restrict__ nbi,
                               const float* __restrict__ gs, float* __restrict__ out, int n) {
  const int i = blockIdx.x * blockDim.x + threadIdx.x;
  if (i >= n) return;
  const int g = nbi[i];
  const float cnt = gs[(size_t)g * 4 + 3];
#pragma unroll
  for (int k = 0; k < 3; ++k)
    out[(size_t)i * 3 + k] = x[(size_t)i * 3 + k] - gs[(size_t)g * 4 + k] / cnt;
}

// ---------------------------------------------------------------------------
// Host orchestration
// ---------------------------------------------------------------------------

static inline int cdiv_(long a, long b) { return (int)((a + b - 1) / b); }

static void rgemm(hipStream_t st, const float* A0, int lda0, int K0,
                  const float* A1, int lda1, int K1,
                  const float* W, const float* b, float* Y, int ldy,
                  int M, int Nc, int act) {
  const float* A1s = A1 ? A1 : A0;   // safe dummy pointer when no second segment
  const int lda1s = A1 ? lda1 : 0;
  const int K1s   = A1 ? K1 : 0;
  if (Nc == 64 && ldy == 64 && b) {
    gvpvf_gemm_wmma64<<<cdiv_(M, 16), 32, 0, st>>>(A0, lda0, K0, A1s, lda1s, K1s,
                                                   W, b, Y, M, act);
  } else {
    dim3 grid(cdiv_(M, 16), cdiv_(Nc, 16));
    gvpvf_gemm_wmma<<<grid, 32, 0, st>>>(A0, lda0, K0, A1s, lda1s, K1s,
                                         W, b, Y, ldy, M, Nc, act);
  }
}

struct GVPP { const float *Wh, *Wu, *fW, *fb, *gW, *gb; };
struct ConvP { const float *ln1b, *ln1g, *ln2b, *ln2g; GVPP msg[3], upd[3]; };

extern "C" void kernel_launch(void* const* d_in, const int* in_sizes, int n_in,
                              void* d_out, int out_size, void* d_ws, size_t ws_size,
                              hipStream_t stream) {
  const int N = 32000, E = 320000, Eu = 160000, G = 800;
  (void)in_sizes; (void)n_in; (void)out_size; (void)ws_size;

  const float* a_t = (const float*)d_in[0];
  const float* c_t = (const float*)d_in[1];
  const float* x_t = (const float*)d_in[2];
  const float* e_t = (const float*)d_in[3];
  const float* t_  = (const float*)d_in[4];
  const int* nbi   = (const int*)d_in[5];
  const int* src   = (const int*)d_in[6];
  const int* dst   = (const int*)d_in[7];
  // d_in[8] = upper_edge_mask (unused by the reference forward)

  // ---- params: jax pytree order (dict keys sorted alphabetically) ----
  int pi = 9;
  auto P = [&]() -> const float* { return (const float*)d_in[pi++]; };
  ConvP cv[4];
  for (int c = 0; c < 4; ++c) {
    cv[c].ln1b = P(); cv[c].ln1g = P();
    cv[c].ln2b = P(); cv[c].ln2g = P();
    for (int m = 0; m < 3; ++m) { GVPP& g = cv[c].msg[m]; g.Wh = P(); g.Wu = P(); g.fW = P(); g.fb = P(); g.gW = P(); g.gb = P(); }
    for (int m = 0; m < 3; ++m) { GVPP& g = cv[c].upd[m]; g.Wh = P(); g.Wu = P(); g.fW = P(); g.fb = P(); g.gW = P(); g.gb = P(); }
  }
  const float *ee_l1W = P(), *ee_l1b = P(), *ee_l2W = P(), *ee_l2b = P(), *ee_lnb = P(), *ee_lng = P();
  const float *eh_l1W = P(), *eh_l1b = P(), *eh_l2W = P(), *eh_l2b = P();
  const float *eu_l1W = P(), *eu_l1b = P(), *eu_l2W = P(), *eu_l2b = P(), *eu_lnb = P(), *eu_lng = P();
  const float *nh_l1W = P(), *nh_l1b = P(), *nh_l2W = P(), *nh_l2b = P();
  GVPP pu[3];
  for (int m = 0; m < 3; ++m) { pu[m].Wh = P(); pu[m].Wu = P(); pu[m].fW = P(); pu[m].fb = P(); pu[m].gW = P(); pu[m].gb = P(); }
  const float *se_l1W = P(), *se_l1b = P(), *se_l2W = P(), *se_l2b = P(), *se_lnb = P(), *se_lng = P();

  // ---- workspace carve (fp32) ----
  float* w = (float*)d_ws;
  auto A = [&](size_t nfl) -> float* { float* r = w; w += nfl; return r; };
  float* S_   = A((size_t)N * 64);
  float* Vv   = A((size_t)N * 48);
  float* Ee   = A((size_t)E * 64);
  float* X    = A((size_t)N * 3);
  float* XD   = A((size_t)E * 3);
  float* RB   = A((size_t)E * 16);
  float* EB0  = A((size_t)E * 208);   // edge scalar concat (also reused for pair bufs)
  float* EB1  = A((size_t)E * 64);
  float* EB2  = A((size_t)E * 64);
  float* MV   = A((size_t)E * 51);    // current edge vector features
  float* VU   = A((size_t)E * 51);    // Vu scratch
  float* SHb  = A((size_t)E * 17);
  float* GTb  = A((size_t)E * 16);
  float* MSUM = A((size_t)N * 64);
  float* VSUM = A((size_t)N * 48);
  float* NB17 = A((size_t)N * 17);
  float* NBa  = A((size_t)N * 64);
  float* NBb  = A((size_t)N * 64);
  float* NVa  = A((size_t)N * 48);
  float* NVb  = A((size_t)N * 48);
  float* NSH  = A((size_t)N * 16);
  float* NGT  = A((size_t)N * 16);
  float* NB16 = A((size_t)N * 16);
  float* GS   = A((size_t)G * 4);
  float* PB  = EB0;                   // Eu x 64 (aliases EB0, safe at head stage)
  float* PB2 = EB0 + (size_t)Eu * 64;

  float* out = (float*)d_out;
  float* out_x = out;                       // N*3
  float* out_c = out + (size_t)N * 3;       // N*6
  float* out_a = out_c + (size_t)N * 6;     // N*10
  float* out_e = out_a + (size_t)N * 10;    // Eu*5

  const int TB = 256;

  // ---- init ----
  hipMemcpyAsync(X, x_t, (size_t)N * 3 * sizeof(float), hipMemcpyDeviceToDevice, stream);
  hipMemsetAsync(Vv, 0, (size_t)N * 48 * sizeof(float), stream);

  // ---- scalar embedding ----
  gvpvf_build_s0<<<cdiv_(N, TB), TB, 0, stream>>>(a_t, c_t, t_, nbi, NB17, N);
  rgemm(stream, NB17, 17, 17, nullptr, 0, 0, se_l1W, se_l1b, NBa, 64, N, 64, ACT_SILU);
  rgemm(stream, NBa, 64, 64, nullptr, 0, 0, se_l2W, se_l2b, NBb, 64, N, 64, ACT_SILU);
  gvpvf_ln64<<<cdiv_(N, 128), 128, 0, stream>>>(NBb, nullptr, 0.f, se_lng, se_lnb, S_, N);

  // ---- edge embedding ----
  rgemm(stream, e_t, 5, 5, nullptr, 0, 0, ee_l1W, ee_l1b, EB1, 64, E, 64, ACT_SILU);
  rgemm(stream, EB1, 64, 64, nullptr, 0, 0, ee_l2W, ee_l2b, EB2, 64, E, 64, ACT_SILU);
  gvpvf_ln64<<<cdiv_(E, 128), 128, 0, stream>>>(EB2, nullptr, 0.f, ee_lng, ee_lnb, Ee, E);

  // ---- initial distances ----
  gvpvf_distances<<<cdiv_(E, TB), TB, 0, stream>>>(X, src, dst, XD, RB, E);

  for (int ci = 0; ci < 4; ++ci) {
    const ConvP& cp = cv[ci];
    // ---- messages over edges ----
    gvpvf_gather_msg<<<cdiv_(E, TB), TB, 0, stream>>>(S_, Vv, Ee, RB, XD, src, dst, EB0, MV, E);
    // GVP 0 (vin=17,h=17,vout=16)
    gvpvf_gvp_vec<17, 17, 16><<<cdiv_(E, 128), 128, 0, stream>>>(MV, cp.msg[0].Wh, cp.msg[0].Wu, SHb, VU, E);
    rgemm(stream, EB0, 208, 208, SHb, 17, 17, cp.msg[0].fW, cp.msg[0].fb, EB1, 64, E, 64, ACT_SILU);
    rgemm(stream, EB1, 64, 64, nullptr, 0, 0, cp.msg[0].gW, cp.msg[0].gb, GTb, 16, E, 16, ACT_SIGMOID);
    gvpvf_gate_combine<<<cdiv_((long)E * 48, TB), TB, 0, stream>>>(GTb, VU, MV, E, 16);
    // GVP 1
    gvpvf_gvp_vec<16, 16, 16><<<cdiv_(E, 128), 128, 0, stream>>>(MV, cp.msg[1].Wh, cp.msg[1].Wu, SHb, VU, E);
    rgemm(stream, EB1, 64, 64, SHb, 16, 16, cp.msg[1].fW, cp.msg[1].fb, EB2, 64, E, 64, ACT_SILU);
    rgemm(stream, EB2, 64, 64, nullptr, 0, 0, cp.msg[1].gW, cp.msg[1].gb, GTb, 16, E, 16, ACT_SIGMOID);
    gvpvf_gate_combine<<<cdiv_((long)E * 48, TB), TB, 0, stream>>>(GTb, VU, MV, E, 16);
    // GVP 2
    gvpvf_gvp_vec<16, 16, 16><<<cdiv_(E, 128), 128, 0, stream>>>(MV, cp.msg[2].Wh, cp.msg[2].Wu, SHb, VU, E);
    rgemm(stream, EB2, 64, 64, SHb, 16, 16, cp.msg[2].fW, cp.msg[2].fb, EB1, 64, E, 64, ACT_SILU);
    rgemm(stream, EB1, 64, 64, nullptr, 0, 0, cp.msg[2].gW, cp.msg[2].gb, GTb, 16, E, 16, ACT_SIGMOID);
    gvpvf_gate_combine<<<cdiv_((long)E * 48, TB), TB, 0, stream>>>(GTb, VU, MV, E, 16);

    // ---- aggregate to nodes ----
    hipMemsetAsync(MSUM, 0, (size_t)N * 64 * sizeof(float), stream);
    hipMemsetAsync(VSUM, 0, (size_t)N * 48 * sizeof(float), stream);
    gvpvf_scatter_add<<<cdiv_((long)E * 64, TB), TB, 0, stream>>>(EB1, dst, MSUM, E, 64);
    gvpvf_scatter_add<<<cdiv_((long)E * 48, TB), TB, 0, stream>>>(MV, dst, VSUM, E, 48);
    gvpvf_ln64<<<cdiv_(N, 128), 128, 0, stream>>>(S_, MSUM, 0.01f, cp.ln1g, cp.ln1b, S_, N);
    gvpvf_add_scaled<<<cdiv_((long)N * 48, TB), TB, 0, stream>>>(Vv, VSUM, 0.01f, (size_t)N * 48);

    // ---- node update GVP chain (us0=S_, uv0=Vv) ----
    gvpvf_gvp_vec<16, 16, 16><<<cdiv_(N, 128), 128, 0, stream>>>(Vv, cp.upd[0].Wh, cp.upd[0].Wu, NSH, NVb, N);
    rgemm(stream, S_, 64, 64, NSH, 16, 16, cp.upd[0].fW, cp.upd[0].fb, NBa, 64, N, 64, ACT_SILU);
    rgemm(stream, NBa, 64, 64, nullptr, 0, 0, cp.upd[0].gW, cp.upd[0].gb, NGT, 16, N, 16, ACT_SIGMOID);
    gvpvf_gate_combine<<<cdiv_((long)N * 48, TB), TB, 0, stream>>>(NGT, NVb, NVa, N, 16);

    gvpvf_gvp_vec<16, 16, 16><<<cdiv_(N, 128), 128, 0, stream>>>(NVa, cp.upd[1].Wh, cp.upd[1].Wu, NSH, NVb, N);
    rgemm(stream, NBa, 64, 64, NSH, 16, 16, cp.upd[1].fW, cp.upd[1].fb, NBb, 64, N, 64, ACT_SILU);
    rgemm(stream, NBb, 64, 64, nullptr, 0, 0, cp.upd[1].gW, cp.upd[1].gb, NGT, 16, N, 16, ACT_SIGMOID);
    gvpvf_gate_combine<<<cdiv_((long)N * 48, TB), TB, 0, stream>>>(NGT, NVb, NVa, N, 16);

    gvpvf_gvp_vec<16, 16, 16><<<cdiv_(N, 128), 128, 0, stream>>>(NVa, cp.upd[2].Wh, cp.upd[2].Wu, NSH, NVb, N);
    rgemm(stream, NBb, 64, 64, NSH, 16, 16, cp.upd[2].fW, cp.upd[2].fb, NBa, 64, N, 64, ACT_SILU);
    rgemm(stream, NBa, 64, 64, nullptr, 0, 0, cp.upd[2].gW, cp.upd[2].gb, NGT, 16, N, 16, ACT_SIGMOID);
    gvpvf_gate_combine<<<cdiv_((long)N * 48, TB), TB, 0, stream>>>(NGT, NVb, NVa, N, 16);

    gvpvf_ln64<<<cdiv_(N, 128), 128, 0, stream>>>(S_, NBa, 1.0f, cp.ln2g, cp.ln2b, S_, N);
    gvpvf_add_scaled<<<cdiv_((long)N * 48, TB), TB, 0, stream>>>(Vv, NVa, 1.0f, (size_t)N * 48);

    // ---- position / edge updates after conv 1 and conv 3 ----
    if (ci == 1 || ci == 3) {
      gvpvf_gvp_vec<16, 16, 16><<<cdiv_(N, 128), 128, 0, stream>>>(Vv, pu[0].Wh, pu[0].Wu, NSH, NVb, N);
      rgemm(stream, S_, 64, 64, NSH, 16, 16, pu[0].fW, pu[0].fb, NBa, 64, N, 64, ACT_SILU);
      rgemm(stream, NBa, 64, 64, nullptr, 0, 0, pu[0].gW, pu[0].gb, NGT, 16, N, 16, ACT_SIGMOID);
      gvpvf_gate_combine<<<cdiv_((long)N * 48, TB), TB, 0, stream>>>(NGT, NVb, NVa, N, 16);

      gvpvf_gvp_vec<16, 16, 16><<<cdiv_(N, 128), 128, 0, stream>>>(NVa, pu[1].Wh, pu[1].Wu, NSH, NVb, N);
      rgemm(stream, NBa, 64, 64, NSH, 16, 16, pu[1].fW, pu[1].fb, NBb, 64, N, 64, ACT_SILU);
      rgemm(stream, NBb, 64, 64, nullptr, 0, 0, pu[1].gW, pu[1].gb, NGT, 16, N, 16, ACT_SIGMOID);
      gvpvf_gate_combine<<<cdiv_((long)N * 48, TB), TB, 0, stream>>>(NGT, NVb, NVa, N, 16);

      gvpvf_gvp_vec<16, 16, 1><<<cdiv_(N, 128), 128, 0, stream>>>(NVa, pu[2].Wh, pu[2].Wu, NSH, NVb, N);
      rgemm(stream, NBb, 64, 64, NSH, 16, 16, pu[2].fW, pu[2].fb, NBa, 64, N, 64, ACT_SILU);
      rgemm(stream, NBa, 64, 64, nullptr, 0, 0, pu[2].gW, pu[2].gb, NGT, 1, N, 1, ACT_NONE);
      gvpvf_gate_combine<<<cdiv_((long)N * 3, TB), TB, 0, stream>>>(NGT, NVb, NVa, N, 1);
      gvpvf_add_scaled<<<cdiv_((long)N * 3, TB), TB, 0, stream>>>(X, NVa, 1.0f, (size_t)N * 3);

      // edge update
      gvpvf_gather_eupd<<<cdiv_(E, TB), TB, 0, stream>>>(S_, Ee, src, dst, EB0, E);
      rgemm(stream, EB0, 192, 192, nullptr, 0, 0, eu_l1W, eu_l1b, EB1, 64, E, 64, ACT_SILU);
      rgemm(stream, EB1, 64, 64, nullptr, 0, 0, eu_l2W, eu_l2b, EB2, 64, E, 64, ACT_SILU);
      gvpvf_ln64<<<cdiv_(E, 128), 128, 0, stream>>>(Ee, EB2, 1.0f, eu_lng, eu_lnb, Ee, E);

      if (ci != 3)
        gvpvf_distances<<<cdiv_(E, TB), TB, 0, stream>>>(X, src, dst, XD, RB, E);
    }
  }

  // ---- node head ----
  rgemm(stream, S_, 64, 64, nullptr, 0, 0, nh_l1W, nh_l1b, NBa, 64, N, 64, ACT_SILU);
  rgemm(stream, NBa, 64, 64, nullptr, 0, 0, nh_l2W, nh_l2b, NB16, 16, N, 16, ACT_NONE);
  gvpvf_head_split<<<cdiv_(N, TB), TB, 0, stream>>>(NB16, out_c, out_a, N);

  // ---- edge head ----
  gvpvf_pair_add<<<cdiv_((long)Eu * 64, TB), TB, 0, stream>>>(Ee, PB, (size_t)Eu * 64, (size_t)Eu * 64);
  rgemm(stream, PB, 64, 64, nullptr, 0, 0, eh_l1W, eh_l1b, PB2, 64, Eu, 64, ACT_SILU);
  rgemm(stream, PB2, 64, 64, nullptr, 0, 0, eh_l2W, eh_l2b, out_e, 5, Eu, 5, ACT_NONE);

  // ---- center positions per graph ----
  hipMemsetAsync(GS, 0, (size_t)G * 4 * sizeof(float), stream);
  gvpvf_graph_accum<<<cdiv_(N, TB), TB, 0, stream>>>(X, nbi, GS, N);
  gvpvf_center_x<<<cdiv_(N, TB), TB, 0, stream>>>(X, nbi, GS, out_x, N);
}